// GroupDivider_70231305224195
// MI455X (gfx1250) — compile-verified
//
#include <hip/hip_runtime.h>

// GroupDivider for MI455X (gfx1250, wave32).
// B=32 batches, N=8192 points, G=256 FPS centers, K=32 nearest neighbors.
//
// Roofline: ~1 GFLOP total, 3 MB unique input -> latency/VALU bound, not HBM
// bound (23.3 TB/s). Strategy: LDS-resident point cloud (96KB/batch fits in
// 320KB WGP LDS) for FPS; WMMA F32 16x16x4 for the pairwise distance phase
// (A row = (-2cx,-2cy,-2cz,|c|^2), B col = (px,py,pz,1) => |c|^2 - 2 c.p,
// + per-column |p|^2 afterwards); wave32 ballot/shuffle register top-32
// (one candidate per lane) for k-NN selection; LDS chunking avoids a 256MB
// global distance matrix. The compute phase is split into a preload pass
// (8 independent global_load_b96 in flight) and a WMMA/store pass so load
// latency overlaps instead of serializing on s_wait_loadcnt 0x0 per tile.

#define B_ 32
#define N_ 8192
#define G_ 256
#define K_ 32
#define CHUNK 2048
#define PITCH (CHUNK + 4)       // pad to break LDS bank alignment across rows
#define TILES_PER_WAVE ((CHUNK / 16) / 16)   // 8

typedef __attribute__((ext_vector_type(2))) float v2f;
typedef __attribute__((ext_vector_type(8))) float v8f;

// ---------------------------------------------------------------------------
// Kernel 1: farthest point sampling. One workgroup per batch, 256 threads.
// LDS: x[N], y[N], z[N], dist[N] (128KB) + reduction scratch.
// ---------------------------------------------------------------------------
__global__ void fps_kernel(const float* __restrict__ xyz,
                           float* __restrict__ center_out) {
  extern __shared__ char smem[];
  float* lx    = (float*)smem;
  float* ly    = lx + N_;
  float* lz    = ly + N_;
  float* ldist = lz + N_;
  float* red_d = ldist + N_;
  int*   red_i = (int*)(red_d + 256);
  int*   s_far = red_i + 256;

  const int b   = blockIdx.x;
  const int tid = threadIdx.x;
  const float* base = xyz + (size_t)b * N_ * 3;

  for (int n = tid; n < N_; n += 256) {
    lx[n] = base[n * 3 + 0];
    ly[n] = base[n * 3 + 1];
    lz[n] = base[n * 3 + 2];
    ldist[n] = 1e10f;
  }
  if (tid == 0) s_far[0] = 0;   // deterministic start at index 0 (matches ref)
  __syncthreads();

  for (int it = 0; it < G_; ++it) {
    const int far = s_far[0];
    const float fx = lx[far], fy = ly[far], fz = lz[far];
    if (tid == 0) {  // emit centroid used in this step (ref scan semantics)
      float* c = center_out + ((size_t)b * G_ + it) * 3;
      c[0] = fx; c[1] = fy; c[2] = fz;
    }
    float bestd = -1.0f;
    int   besti = 0;
    for (int n = tid; n < N_; n += 256) {
      float dx = lx[n] - fx, dy = ly[n] - fy, dz = lz[n] - fz;
      float d  = dx * dx + dy * dy + dz * dz;
      float nd = fminf(ldist[n], d);
      ldist[n] = nd;
      if (nd > bestd) { bestd = nd; besti = n; }
    }
    red_d[tid] = bestd;
    red_i[tid] = besti;
    __syncthreads();
    for (int s = 128; s > 0; s >>= 1) {
      if (tid < s) {
        float d2 = red_d[tid + s]; int i2 = red_i[tid + s];
        float d1 = red_d[tid];     int i1 = red_i[tid];
        if (d2 > d1 || (d2 == d1 && i2 < i1)) { red_d[tid] = d2; red_i[tid] = i2; }
      }
      __syncthreads();
    }
    if (tid == 0) s_far[0] = red_i[0];
    __syncthreads();
  }
}

// ---------------------------------------------------------------------------
// Kernel 2: WMMA pairwise distances + wave-per-center top-32 + grouped mean.
// Grid: B*16 workgroups (16 centers each), 512 threads = 16 wave32 waves.
// LDS: A-coeffs sA[16*4] + distance chunk sD[16][PITCH] (~128KB).
// ---------------------------------------------------------------------------
__global__ void knn_kernel(const float* __restrict__ xyz,
                           const float* __restrict__ center,
                           float* __restrict__ feat_out) {
  extern __shared__ char smem[];
  float* sA = (float*)smem;      // 16 rows * (-2x,-2y,-2z,|c|^2)
  float* sD = sA + 64;           // 16 * PITCH distance chunk

  const int b    = blockIdx.x >> 4;
  const int cg   = blockIdx.x & 15;         // center-group within batch
  const int tid  = threadIdx.x;
  const int w    = tid >> 5;                // wave id: 0..15
  const int lane = tid & 31;
  const float* pbase = xyz + (size_t)b * N_ * 3;

  if (tid < 16) {
    const float* c = center + ((size_t)b * G_ + cg * 16 + tid) * 3;
    float x = c[0], y = c[1], z = c[2];
    sA[tid * 4 + 0] = -2.0f * x;
    sA[tid * 4 + 1] = -2.0f * y;
    sA[tid * 4 + 2] = -2.0f * z;
    sA[tid * 4 + 3] = x * x + y * y + z * z;
  }
  __syncthreads();

  // A fragment (16x4 f32): lanes 0-15 hold K=0,1; lanes 16-31 hold K=2,3.
  const int  m  = lane & 15;
  const bool hi = lane >= 16;
  v2f afrag;
  afrag[0] = hi ? sA[m * 4 + 2] : sA[m * 4 + 0];
  afrag[1] = hi ? sA[m * 4 + 3] : sA[m * 4 + 1];

  // Register top-32 list: one (dist,idx) entry per lane of the wave.
  float topd = 1e10f;
  int   topi = 0;
  float thr  = 1e10f;   // current 32nd-smallest (wave max of topd)

  for (int chunk = 0; chunk < N_ / CHUNK; ++chunk) {
    const int cbase = chunk * CHUNK;

    // Prefetch next chunk's coordinates (24KB) while we crunch this one.
    if (chunk + 1 < N_ / CHUNK) {
      const char* np = (const char*)(pbase + (size_t)(cbase + CHUNK) * 3);
      __builtin_prefetch(np + (size_t)tid * 48, 0, 1);
    }

    // ---- compute phase: 128 WMMA tiles per chunk, 8 per wave. ----
    // Pass 1: preload all 8 tile points -> 8 independent b96 loads in flight.
    float qx[TILES_PER_WAVE], qy[TILES_PER_WAVE], qz[TILES_PER_WAVE];
#pragma unroll
    for (int tt = 0; tt < TILES_PER_WAVE; ++tt) {
      const int col = (tt * 16 + w) * 16 + m;       // column within chunk
      const float* p = pbase + (size_t)(cbase + col) * 3;
      qx[tt] = p[0];
      qy[tt] = p[1];
      qz[tt] = p[2];
    }
    // Pass 2: WMMA each tile and commit squared distances to LDS.
#pragma unroll
    for (int tt = 0; tt < TILES_PER_WAVE; ++tt) {
      const int col = (tt * 16 + w) * 16 + m;
      const float px = qx[tt], py = qy[tt], pz = qz[tt];
      const float pp = px * px + py * py + pz * pz;
      // B fragment (4x16 f32): lanes 0-15 rows K=0,1; lanes 16-31 rows K=2,3.
      v2f bfrag;
      bfrag[0] = hi ? pz : px;
      bfrag[1] = hi ? 1.0f : py;
      v8f acc = {};
      acc = __builtin_amdgcn_wmma_f32_16x16x4_f32(
          /*neg_a=*/false, afrag, /*neg_b=*/false, bfrag,
          /*c_mod=*/(short)0, acc, /*reuse_a=*/false, /*reuse_b=*/false);
      // D layout: VGPR v -> M = v + 8*(lane/16), N = lane%16.
      const int mbase = hi ? 8 : 0;
#pragma unroll
      for (int v = 0; v < 8; ++v)
        sD[(mbase + v) * PITCH + col] = acc[v] + pp;   // full sq distance
    }
    __syncthreads();

    // ---- select phase: wave w scans center-row w of the chunk ----
    for (int t = 0; t < CHUNK; t += 32) {
      const float d  = sD[w * PITCH + t + lane];
      const int   gi = cbase + t + lane;
      unsigned cand = (unsigned)__ballot(d < thr);
      while (cand) {
        const int src = __ffs(cand) - 1;
        cand &= cand - 1;
        const float dc = __shfl(d, src);
        const int   ic = __shfl(gi, src);
        if (dc < thr) {            // may have shrunk since the ballot
          // evict current wave-max entry
          float mx = topd;
#pragma unroll
          for (int s = 16; s >= 1; s >>= 1) mx = fmaxf(mx, __shfl_xor(mx, s));
          const unsigned winb = (unsigned)__ballot(topd == mx);
          const int winner = __ffs(winb) - 1;
          if (lane == winner) { topd = dc; topi = ic; }
          float nm = topd;
#pragma unroll
          for (int s = 16; s >= 1; s >>= 1) nm = fmaxf(nm, __shfl_xor(nm, s));
          thr = nm;
        }
      }
    }
    __syncthreads();
  }

  // ---- gather the 32 nearest (one per lane) and mean-reduce ----
  const float* p = pbase + (size_t)topi * 3;
  float sx = p[0], sy = p[1], sz = p[2];
#pragma unroll
  for (int s = 16; s >= 1; s >>= 1) {
    sx += __shfl_xor(sx, s);
    sy += __shfl_xor(sy, s);
    sz += __shfl_xor(sz, s);
  }
  if (lane == 0) {
    float* f = feat_out + ((size_t)b * G_ + cg * 16 + w) * 3;
    const float inv = 1.0f / (float)K_;
    f[0] = sx * inv; f[1] = sy * inv; f[2] = sz * inv;
  }
}

// ---------------------------------------------------------------------------
extern "C" void kernel_launch(void* const* d_in, const int* in_sizes, int n_in,
                              void* d_out, int out_size, void* d_ws, size_t ws_size,
                              hipStream_t stream) {
  (void)in_sizes; (void)n_in; (void)out_size; (void)d_ws; (void)ws_size;
  const float* xyz = (const float*)d_in[0];
  float* out        = (float*)d_out;
  float* center_out = out;                          // (B,G,3) first
  float* feat_out   = out + (size_t)B_ * G_ * 3;    // (B,G,3) second

  const size_t smem1 = (size_t)(4 * N_) * sizeof(float)   // x,y,z,dist
                     + 256 * sizeof(float) + 256 * sizeof(int) + 16;
  const size_t smem2 = 64 * sizeof(float) + (size_t)16 * PITCH * sizeof(float);

  // Allow >64KB dynamic LDS (CDNA5 WGP has 320KB). Idempotent, graph-safe.
  (void)hipFuncSetAttribute(reinterpret_cast<const void*>(fps_kernel),
                            hipFuncAttributeMaxDynamicSharedMemorySize, (int)smem1);
  (void)hipFuncSetAttribute(reinterpret_cast<const void*>(knn_kernel),
                            hipFuncAttributeMaxDynamicSharedMemorySize, (int)smem2);

  fps_kernel<<<dim3(B_), dim3(256), smem1, stream>>>(xyz, center_out);
  knn_kernel<<<dim3(B_ * 16), dim3(512), smem2, stream>>>(xyz, center_out, feat_out);
}